// JKNet_12343736009310
// MI455X (gfx1250) — compile-verified
//
#include <hip/hip_runtime.h>
#include <hip/hip_bf16.h>

typedef __attribute__((ext_vector_type(2))) float v2f;
typedef __attribute__((ext_vector_type(8))) float v8f;

__device__ __forceinline__ v8f wmma_f32_k4(v2f a, v2f b, v8f c) {
    // D = A(16x4,f32) * B(4x16,f32) + C(16x16,f32)
    return __builtin_amdgcn_wmma_f32_16x16x4_f32(
        /*neg_a=*/false, a, /*neg_b=*/false, b,
        /*c_mod=*/(short)0, c, /*reuse_a=*/false, /*reuse_b=*/false);
}

// ---------------------------------------------------------------- degree ----
__global__ __launch_bounds__(256) void k_init_deg(float* __restrict__ deg, int n) {
    int i = blockIdx.x * 256 + threadIdx.x;
    if (i < n) deg[i] = 1.0f;  // self-loop
}

__global__ __launch_bounds__(256) void k_deg_accum(float* __restrict__ deg,
                                                   const int* __restrict__ dst, int e) {
    int i = blockIdx.x * 256 + threadIdx.x;
    if (i < e) atomicAdd(deg + dst[i], 1.0f);
}

__global__ __launch_bounds__(256) void k_rsqrt(float* __restrict__ deg, int n) {
    int i = blockIdx.x * 256 + threadIdx.x;
    if (i < n) deg[i] = rsqrtf(deg[i]);   // deg becomes dinv
}

// ------------------------------------------------------- layer GEMM (H@W) ---
// One wave computes a 16x64 output strip with 4 f32 WMMA accumulators.
// W staged in LDS pair-interleaved: sW[(k/2)*128 + n*2 + (k&1)] = W[k][n],
// so each B operand is one aligned ds_load_b64 into an even VGPR pair.
// Also zeroes the matching rows of AGG for the upcoming scatter.
__global__ __launch_bounds__(256) void k_gemm64(const float* __restrict__ A, int lda,
                                                const float* __restrict__ W,   // 64x64
                                                float* __restrict__ G,
                                                float* __restrict__ AGG, int nrows) {
    __shared__ float sW[64 * 64];
    for (int i = threadIdx.x; i < 64 * 64; i += 256) {
        int k = i >> 6, n = i & 63;
        sW[(k >> 1) * 128 + n * 2 + (k & 1)] = W[i];
    }
    __syncthreads();

    const int lane  = threadIdx.x & 31;
    const int wave  = threadIdx.x >> 5;
    const int strip = blockIdx.x * 8 + wave;
    const int r0    = strip * 16;
    if (r0 >= nrows) return;               // wave-uniform: EXEC stays all-1s

    const int m  = lane & 15;              // row-in-tile (A) / col-in-tile (B,D)
    const int kh = lane >> 4;              // K half-select per ISA A/B layout

    v8f acc0 = {}, acc1 = {}, acc2 = {}, acc3 = {};
    int arow = r0 + m; if (arow >= nrows) arow = nrows - 1;
    const float* ap = A + (size_t)arow * lda + kh * 2;

#pragma unroll
    for (int k0 = 0; k0 < 64; k0 += 4) {
        float2 av = *(const float2*)(ap + k0);
        v2f a; a.x = av.x; a.y = av.y;
        // pair index = (k0 + kh*2)/2 = k0/2 + kh ; row stride 128 floats
        const float* wb = sW + ((k0 >> 1) + kh) * 128 + m * 2;
        acc0 = wmma_f32_k4(a, *(const v2f*)(wb +  0), acc0);  // n-tile 0
        acc1 = wmma_f32_k4(a, *(const v2f*)(wb + 32), acc1);  // n-tile 1
        acc2 = wmma_f32_k4(a, *(const v2f*)(wb + 64), acc2);  // n-tile 2
        acc3 = wmma_f32_k4(a, *(const v2f*)(wb + 96), acc3);  // n-tile 3
    }

#pragma unroll
    for (int v = 0; v < 8; ++v) {
        int row = r0 + v + kh * 8;         // C/D layout: M = v + 8*(lane>=16)
        if (row < nrows) {
            float* gp = G   + (size_t)row * 64 + m;
            float* zp = AGG + (size_t)row * 64 + m;
            gp[0] = acc0[v]; gp[16] = acc1[v]; gp[32] = acc2[v]; gp[48] = acc3[v];
            zp[0] = 0.0f;    zp[16] = 0.0f;    zp[32] = 0.0f;    zp[48] = 0.0f;
        }
    }
}

// ------------------------------------------------------------- edge scatter -
// 16 threads per edge, one float4 feature chunk each; L2-resident atomics.
__global__ __launch_bounds__(256) void k_scatter(const float* __restrict__ G,
                                                 float* __restrict__ AGG,
                                                 const float* __restrict__ dinv,
                                                 const int* __restrict__ src,
                                                 const int* __restrict__ dst, int e) {
    long long t = (long long)blockIdx.x * 256 + threadIdx.x;
    if (t >= (long long)e * 16) return;
    int ed = (int)(t >> 4);
    int q  = ((int)t & 15) * 4;
    int s  = src[ed], d = dst[ed];
    float w = dinv[s] * dinv[d];
    float4 gv = *(const float4*)(G + (size_t)s * 64 + q);
    float* apd = AGG + (size_t)d * 64 + q;
    atomicAdd(apd + 0, gv.x * w);
    atomicAdd(apd + 1, gv.y * w);
    atomicAdd(apd + 2, gv.z * w);
    atomicAdd(apd + 3, gv.w * w);
}

// ------------------------------------------- h = relu(agg + g*dinv^2 + b) ---
// Writes directly into the JK concat buffer (row stride 256).
__global__ __launch_bounds__(256) void k_update(const float* __restrict__ AGG,
                                                const float* __restrict__ G,
                                                const float* __restrict__ dinv,
                                                const float* __restrict__ bias,
                                                float* __restrict__ Hout,  // jk + l*64
                                                int n) {
    int t = blockIdx.x * 256 + threadIdx.x;
    if (t >= n * 16) return;
    int node = t >> 4;
    int q    = (t & 15) * 4;
    float ns = dinv[node] * dinv[node];
    float4 av = *(const float4*)(AGG + (size_t)node * 64 + q);
    float4 gv = *(const float4*)(G   + (size_t)node * 64 + q);
    float4 bv = *(const float4*)(bias + q);
    float4 h;
    h.x = fmaxf(av.x + gv.x * ns + bv.x, 0.0f);
    h.y = fmaxf(av.y + gv.y * ns + bv.y, 0.0f);
    h.z = fmaxf(av.z + gv.z * ns + bv.z, 0.0f);
    h.w = fmaxf(av.w + gv.w * ns + bv.w, 0.0f);
    *(float4*)(Hout + (size_t)node * 256 + q) = h;
}

// --------------------------------------------- final GEMM: jk@linW + linb ---
// A = jk [n x 256], B = lin_W [256 x 32] pair-interleaved in LDS:
// sW[(k/2)*64 + n*2 + (k&1)] = LW[k][n]. One wave -> 16x32 strip, 2 acc tiles.
__global__ __launch_bounds__(256) void k_gemm_final(const float* __restrict__ JK,
                                                    const float* __restrict__ LW,
                                                    const float* __restrict__ LB,
                                                    float* __restrict__ OUT, int nrows) {
    __shared__ float sW[256 * 32];   // 32 KB LDS
    for (int i = threadIdx.x; i < 256 * 32; i += 256) {
        int k = i >> 5, n = i & 31;
        sW[(k >> 1) * 64 + n * 2 + (k & 1)] = LW[i];
    }
    __syncthreads();

    const int lane  = threadIdx.x & 31;
    const int wave  = threadIdx.x >> 5;
    const int strip = blockIdx.x * 8 + wave;
    const int r0    = strip * 16;
    if (r0 >= nrows) return;

    const int m  = lane & 15;
    const int kh = lane >> 4;

    v8f acc0 = {}, acc1 = {};
    int arow = r0 + m; if (arow >= nrows) arow = nrows - 1;
    const float* ap = JK + (size_t)arow * 256 + kh * 2;

#pragma unroll 8
    for (int k0 = 0; k0 < 256; k0 += 4) {
        float2 av = *(const float2*)(ap + k0);
        v2f a; a.x = av.x; a.y = av.y;
        const float* wb = sW + ((k0 >> 1) + kh) * 64 + m * 2;
        acc0 = wmma_f32_k4(a, *(const v2f*)(wb +  0), acc0);  // n-tile 0
        acc1 = wmma_f32_k4(a, *(const v2f*)(wb + 32), acc1);  // n-tile 1
    }

    float b0 = LB[m], b1 = LB[16 + m];
#pragma unroll
    for (int v = 0; v < 8; ++v) {
        int row = r0 + v + kh * 8;
        if (row < nrows) {
            float* op = OUT + (size_t)row * 32 + m;
            op[0]  = acc0[v] + b0;
            op[16] = acc1[v] + b1;
        }
    }
}

// ---------------------------------------------------------------------------
extern "C" void kernel_launch(void* const* d_in, const int* in_sizes, int n_in,
                              void* d_out, int out_size, void* d_ws, size_t ws_size,
                              hipStream_t stream) {
    const float* x    = (const float*)d_in[0];
    const int*   edge = (const int*)d_in[1];
    const float* Ws   = (const float*)d_in[2];
    const float* bs   = (const float*)d_in[3];
    const float* linW = (const float*)d_in[4];
    const float* linb = (const float*)d_in[5];
    float*       out  = (float*)d_out;

    const int N = in_sizes[0] / 64;             // 100000
    const int E = in_sizes[1] / 2;              // 1600000
    const int L = in_sizes[2] / (64 * 64);      // 4

    // workspace layout (floats): dinv | g | agg | jk
    float* ws   = (float*)d_ws;
    size_t off  = 0;
    float* deg  = ws + off; off += ((size_t)N + 255) & ~(size_t)255;
    float* g    = ws + off; off += (size_t)N * 64;
    float* agg  = ws + off; off += (size_t)N * 64;
    float* jk   = ws + off;                     // N x (L*64), row stride 256

    const int ldjk = L * 64;                    // 256

    k_init_deg<<<(N + 255) / 256, 256, 0, stream>>>(deg, N);
    k_deg_accum<<<(E + 255) / 256, 256, 0, stream>>>(deg, edge + E, E);
    k_rsqrt<<<(N + 255) / 256, 256, 0, stream>>>(deg, N);

    const int strips   = (N + 15) / 16;
    const int gemmGrid = (strips + 7) / 8;
    const long long scatThreads = (long long)E * 16;
    const int scatGrid = (int)((scatThreads + 255) / 256);
    const int updGrid  = ((N * 16) + 255) / 256;

    for (int l = 0; l < L; ++l) {
        const float* A = (l == 0) ? x : (jk + (size_t)(l - 1) * 64);
        int lda        = (l == 0) ? 64 : ldjk;
        k_gemm64<<<gemmGrid, 256, 0, stream>>>(A, lda, Ws + (size_t)l * 64 * 64, g, agg, N);
        k_scatter<<<scatGrid, 256, 0, stream>>>(g, agg, deg, edge, edge + E, E);
        k_update<<<updGrid, 256, 0, stream>>>(agg, g, deg, bs + (size_t)l * 64,
                                              jk + (size_t)l * 64, N);
    }

    k_gemm_final<<<gemmGrid, 256, 0, stream>>>(jk, linW, linb, out, N);
}